// CapsuleNetwork_32444182954358
// MI455X (gfx1250) — compile-verified
//
#include <hip/hip_runtime.h>
#include <hip/hip_bf16.h>
#include <math.h>

typedef __attribute__((ext_vector_type(2))) float v2f;
typedef __attribute__((ext_vector_type(4))) float v4f;
typedef __attribute__((ext_vector_type(8))) float v8f;

#if defined(__has_builtin)
#if __has_builtin(__builtin_amdgcn_wmma_f32_16x16x4_f32)
#define HAVE_WMMA_F32X4 1
#endif
#endif

#define B_SZ        64
#define CHANNELS    32
#define IN_UNITS    12800          // 32*20*20
#define JM          64             // 8 primary caps * 8 dims
#define NSPLIT      16
#define KSPLIT      (IN_UNITS / NSPLIT)   // 800

// ---------------------------------------------------------------------------
// Kernel 1: conv 9x9 valid (1->32) + bias + relu -> xf[64][12800]
// One block per batch image; input tile + all weights staged in LDS.
// ---------------------------------------------------------------------------
__global__ __launch_bounds__(256) void conv_relu_k(const float* __restrict__ x,
                                                   const float* __restrict__ w,
                                                   const float* __restrict__ bias,
                                                   float* __restrict__ xf) {
  __shared__ float xs[784];     // 28x28
  __shared__ float ws[2592];    // 32*81
  __shared__ float bs[32];
  const int b = blockIdx.x;
  const int t = threadIdx.x;
  for (int i = t; i < 784; i += 256)  xs[i] = x[b * 784 + i];
  for (int i = t; i < 2592; i += 256) ws[i] = w[i];
  if (t < 32) bs[t] = bias[t];
  __syncthreads();
  for (int o = t; o < IN_UNITS; o += 256) {
    const int c  = o / 400;
    const int yx = o - c * 400;
    const int y  = yx / 20;
    const int xx = yx - y * 20;
    float acc = bs[c];
    const float* wp = ws + c * 81;
#pragma unroll
    for (int i = 0; i < 9; ++i) {
      const float* xr = xs + (y + i) * 28 + xx;
#pragma unroll
      for (int j = 0; j < 9; ++j)
        acc = fmaf(xr[j], wp[i * 9 + j], acc);
    }
    xf[(size_t)b * IN_UNITS + o] = fmaxf(acc, 0.0f);
  }
}

// ---------------------------------------------------------------------------
// Kernel 2: reduce W1[j][k][m][c] over c (32 contiguous f32) -> A2[k][j*8+m].
// 104.9 MB streamed once with non-temporal 128b loads (read-once data;
// keep it out of the way of L2-resident xf/A2).
// ---------------------------------------------------------------------------
__global__ __launch_bounds__(256) void reduce_w1_k(const float* __restrict__ W1,
                                                   float* __restrict__ A2) {
  const int tid = blockIdx.x * 256 + threadIdx.x;     // 0..819199, exact
  const v4f* p = (const v4f*)(W1 + (size_t)tid * 32); // 128B aligned
  float acc = 0.0f;
#pragma unroll
  for (int i = 0; i < 8; ++i) {
    v4f v = __builtin_nontemporal_load(p + i);
    acc += (v.x + v.y) + (v.z + v.w);
  }
  const int j   = tid / (IN_UNITS * 8);
  const int rem = tid - j * (IN_UNITS * 8);
  const int k   = rem >> 3;
  const int m   = rem & 7;
  A2[(size_t)k * JM + j * 8 + m] = acc;
}

// ---------------------------------------------------------------------------
// Kernel 3: GEMM partials with V_WMMA_F32_16X16X4_F32.
//   part[ks][row 0..63][col 0..63] = sum over K-slice of xf[row][k]*A2[k][col]
// One wave per block; each wave owns a 16x64 strip (4 accumulators sharing
// the A fragment each k-step). K split 16 ways for occupancy; reduction is
// done in a fixed order later -> deterministic.
// ---------------------------------------------------------------------------
__global__ __launch_bounds__(32) void gemm_part_k(const float* __restrict__ xf,
                                                  const float* __restrict__ A2,
                                                  float* __restrict__ part) {
  const int mt   = blockIdx.x & 3;       // M tile (16 rows)
  const int ks   = blockIdx.x >> 2;      // K split 0..15
  const int lane = threadIdx.x;
  const int half = lane >> 4;            // 0: K pair {0,1}, 1: K pair {2,3}
  const int l    = lane & 15;
  const int m0   = mt * 16;
  const int k0   = ks * KSPLIT;
  const float* arow = xf + (size_t)(m0 + l) * IN_UNITS;

#if HAVE_WMMA_F32X4
  v8f acc0 = {}, acc1 = {}, acc2 = {}, acc3 = {};
#pragma unroll 2
  for (int k = k0; k < k0 + KSPLIT; k += 4) {
    const int ka = k + 2 * half;
    v2f a;  a.x = arow[ka];  a.y = arow[ka + 1];           // A 16x4 layout
    const float* br0 = A2 + (size_t)ka * JM + l;           // B row K=ka
    const float* br1 = br0 + JM;                           // B row K=ka+1
    v2f b0; b0.x = br0[0];  b0.y = br1[0];
    v2f b1; b1.x = br0[16]; b1.y = br1[16];
    v2f b2; b2.x = br0[32]; b2.y = br1[32];
    v2f b3; b3.x = br0[48]; b3.y = br1[48];
    acc0 = __builtin_amdgcn_wmma_f32_16x16x4_f32(false, a, false, b0, (short)0, acc0, false, false);
    acc1 = __builtin_amdgcn_wmma_f32_16x16x4_f32(false, a, false, b1, (short)0, acc1, false, false);
    acc2 = __builtin_amdgcn_wmma_f32_16x16x4_f32(false, a, false, b2, (short)0, acc2, false, false);
    acc3 = __builtin_amdgcn_wmma_f32_16x16x4_f32(false, a, false, b3, (short)0, acc3, false, false);
  }
  float* dst = part + (size_t)ks * (64 * 64);
#pragma unroll
  for (int r = 0; r < 8; ++r) {
    const int row = m0 + r + 8 * half;                     // C/D layout
    dst[row * 64 +  0 + l] = acc0[r];
    dst[row * 64 + 16 + l] = acc1[r];
    dst[row * 64 + 32 + l] = acc2[r];
    dst[row * 64 + 48 + l] = acc3[r];
  }
#else
  // Scalar fallback mirroring the same output mapping.
  float* dst = part + (size_t)ks * (64 * 64);
#pragma unroll
  for (int tt = 0; tt < 4; ++tt) {
#pragma unroll
    for (int r = 0; r < 8; ++r) {
      const int row = m0 + r + 8 * half;
      const int col = tt * 16 + l;
      float s = 0.0f;
      for (int k = k0; k < k0 + KSPLIT; ++k)
        s = fmaf(xf[(size_t)row * IN_UNITS + k], A2[(size_t)k * JM + col], s);
      dst[row * 64 + col] = s;
    }
  }
#endif
}

// ---------------------------------------------------------------------------
// Kernel 4: fixed-order reduction of K-split partials, fold in the 1/8 from
// softmax(zeros) over 8 primary capsules.
// ---------------------------------------------------------------------------
__global__ __launch_bounds__(256) void reduce_s1_k(const float* __restrict__ part,
                                                   float* __restrict__ s1) {
  const int t = blockIdx.x * 256 + threadIdx.x;   // 0..4095
  float acc = 0.0f;
#pragma unroll
  for (int s = 0; s < NSPLIT; ++s) acc += part[s * 4096 + t];
  s1[t] = acc * 0.125f;
}

// ---------------------------------------------------------------------------
// Kernel 5: squash(v1) -> digit caps transform (W2) -> 3-iter dynamic routing
// -> out[64][10][16]. One block per batch item; everything lives in LDS.
// ---------------------------------------------------------------------------
__global__ __launch_bounds__(256) void caps_route_k(const float* __restrict__ s1,
                                                    const float* __restrict__ W2,
                                                    float* __restrict__ out) {
  __shared__ float sv[64];
  __shared__ float fac[8];
  __shared__ float v1[64];       // [k][c] primary capsule outputs
  __shared__ float u2[1280];     // [j][k][m] 10*8*16
  __shared__ float blog[80];     // routing logits [j][k]
  __shared__ float cc[80];       // coupling coeffs
  __shared__ float sjm[160];     // [j][m]
  __shared__ float facj[10];
  __shared__ float vout[160];
  const int b = blockIdx.x;
  const int t = threadIdx.x;

  if (t < 64) sv[t] = s1[b * 64 + t];
  if (t < 80) blog[t] = 0.0f;
  __syncthreads();
  if (t < 8) {                              // squash over primary dim (8)
    float n2 = 0.0f;
#pragma unroll
    for (int m = 0; m < 8; ++m) { const float v = sv[t * 8 + m]; n2 = fmaf(v, v, n2); }
    fac[t] = sqrtf(n2) / (1.0f + n2);
  }
  __syncthreads();
  if (t < 64) v1[t] = sv[t] * fac[t >> 3];
  __syncthreads();

  // u2[j,k,m] = sum_c W2[j,k,m,c] * v1[k,c]
  for (int idx = t; idx < 1280; idx += 256) {
    const int k = (idx >> 4) & 7;
    const float* wp = W2 + (size_t)idx * 8;
    float acc = 0.0f;
#pragma unroll
    for (int c = 0; c < 8; ++c) acc = fmaf(wp[c], v1[k * 8 + c], acc);
    u2[idx] = acc;
  }

  for (int it = 0; it < 3; ++it) {
    __syncthreads();
    if (t < 8) {                            // softmax over j (10 classes), per k
      float mx = blog[t];
      for (int j = 1; j < 10; ++j) mx = fmaxf(mx, blog[j * 8 + t]);
      float sum = 0.0f;
      for (int j = 0; j < 10; ++j) {
        const float e = expf(blog[j * 8 + t] - mx);
        cc[j * 8 + t] = e; sum += e;
      }
      const float inv = 1.0f / sum;
      for (int j = 0; j < 10; ++j) cc[j * 8 + t] *= inv;
    }
    __syncthreads();
    if (t < 160) {                          // s[j,m] = sum_k c[j,k]*u2[j,k,m]
      const int j = t >> 4, m = t & 15;
      float s = 0.0f;
#pragma unroll
      for (int k = 0; k < 8; ++k) s = fmaf(cc[j * 8 + k], u2[(j * 8 + k) * 16 + m], s);
      sjm[t] = s;
    }
    __syncthreads();
    if (t < 10) {                           // squash over out dim (16)
      float n2 = 0.0f;
#pragma unroll
      for (int m = 0; m < 16; ++m) { const float v = sjm[t * 16 + m]; n2 = fmaf(v, v, n2); }
      facj[t] = sqrtf(n2) / (1.0f + n2);
    }
    __syncthreads();
    if (t < 160) vout[t] = sjm[t] * facj[t >> 4];
    __syncthreads();
    if (it < 2 && t < 80) {                 // b[j,k] += sum_m u2[j,k,m]*v[j,m]
      const int j = t >> 3;
      float d = 0.0f;
#pragma unroll
      for (int m = 0; m < 16; ++m) d = fmaf(u2[t * 16 + m], vout[j * 16 + m], d);
      blog[t] += d;
    }
  }
  __syncthreads();
  if (t < 160) out[b * 160 + t] = vout[t];
}

// ---------------------------------------------------------------------------
extern "C" void kernel_launch(void* const* d_in, const int* in_sizes, int n_in,
                              void* d_out, int out_size, void* d_ws, size_t ws_size,
                              hipStream_t stream) {
  (void)in_sizes; (void)n_in; (void)out_size; (void)ws_size;
  const float* x      = (const float*)d_in[0];
  const float* conv_w = (const float*)d_in[1];
  const float* conv_b = (const float*)d_in[2];
  const float* W1     = (const float*)d_in[3];
  const float* W2     = (const float*)d_in[4];
  float* out = (float*)d_out;

  char* ws = (char*)d_ws;
  float* xf   = (float*)(ws);                         // 64*12800 f32 = 3,276,800 B
  float* A2   = (float*)(ws + 3276800);               // 12800*64 f32 = 3,276,800 B
  float* part = (float*)(ws + 6553600);               // 16*64*64 f32 = 262,144 B
  float* s1   = (float*)(ws + 6815744);               // 64*64 f32    = 16,384 B

  conv_relu_k <<<B_SZ, 256, 0, stream>>>(x, conv_w, conv_b, xf);
  reduce_w1_k <<<3200, 256, 0, stream>>>(W1, A2);
  gemm_part_k <<<4 * NSPLIT, 32, 0, stream>>>(xf, A2, part);
  reduce_s1_k <<<16, 256, 0, stream>>>(part, s1);
  caps_route_k<<<B_SZ, 256, 0, stream>>>(s1, W2, out);
}